// SGC_66073776882321
// MI455X (gfx1250) — compile-verified
//
#include <hip/hip_runtime.h>

// ---------------------------------------------------------------------------
// SGC on MI455X (gfx1250, wave32):
//   deg[i]  = 1 + #edges with dst==i            (matches reference segment_sum)
//   dinv    = rsqrt(deg)
//   hop x3:  y[d] += x[s] * dinv[s]*dinv[d]     (self term fused into init)
//   out     = h @ W + b  via V_WMMA_F32_16X16X4_F32 (one wave per 16x16 tile)
// ---------------------------------------------------------------------------

typedef __attribute__((ext_vector_type(2))) float v2f;
typedef __attribute__((ext_vector_type(8))) float v8f;

#define SGC_N     50000
#define SGC_E     800000
#define SGC_F     256
#define SGC_C     64

// --- degree init: self-loop contributes 1 to every node ---------------------
__global__ void sgc_deg_init(float* __restrict__ deg, int n) {
    int i = blockIdx.x * blockDim.x + threadIdx.x;
    if (i < n) deg[i] = 1.0f;
}

// --- degree accumulate over edge destinations -------------------------------
__global__ void sgc_deg_accum(const long long* __restrict__ dst,
                              float* __restrict__ deg, int nE) {
    int e = blockIdx.x * blockDim.x + threadIdx.x;
    if (e < nE) atomicAdd(&deg[(int)dst[e]], 1.0f);
}

// --- dinv = rsqrt(deg) (deg >= 1 always due to self loops) ------------------
__global__ void sgc_dinv(const float* __restrict__ deg,
                         float* __restrict__ dinv, int n) {
    int i = blockIdx.x * blockDim.x + threadIdx.x;
    if (i < n) dinv[i] = rsqrtf(deg[i]);
}

// --- per-hop init: y[i][f] = x[i][f] * dinv[i]^2  (self-loop term) ----------
__global__ void sgc_self_init(const float* __restrict__ xin,
                              const float* __restrict__ dinv,
                              float* __restrict__ y, int nElem) {
    int i = blockIdx.x * blockDim.x + threadIdx.x;
    if (i < nElem) {
        float di = dinv[i >> 8];          // F == 256
        y[i] = xin[i] * di * di;
    }
}

// --- edge scatter: wave32 per edge, 8 feats/lane, f32 atomics ---------------
__global__ __launch_bounds__(256)
void sgc_hop_edges(const float* __restrict__ xin,
                   float* __restrict__ y,
                   const long long* __restrict__ src,
                   const long long* __restrict__ dst,
                   const float* __restrict__ dinv, int nE) {
    const int lane = threadIdx.x & 31;
    const int e    = blockIdx.x * 8 + (threadIdx.x >> 5);   // 8 waves/block
    if (e >= nE) return;                                    // wave-uniform
    const int   s = (int)src[e];
    const int   d = (int)dst[e];
    const float w = dinv[s] * dinv[d];

    const float* xs = xin + (size_t)s * SGC_F;
    float*       yd = y   + (size_t)d * SGC_F;

    // two coalesced 512B sweeps across the wave
    const float4 a0 = *(const float4*)(xs + lane * 4);
    const float4 a1 = *(const float4*)(xs + 128 + lane * 4);

    atomicAdd(yd + lane * 4 + 0,       a0.x * w);
    atomicAdd(yd + lane * 4 + 1,       a0.y * w);
    atomicAdd(yd + lane * 4 + 2,       a0.z * w);
    atomicAdd(yd + lane * 4 + 3,       a0.w * w);
    atomicAdd(yd + 128 + lane * 4 + 0, a1.x * w);
    atomicAdd(yd + 128 + lane * 4 + 1, a1.y * w);
    atomicAdd(yd + 128 + lane * 4 + 2, a1.z * w);
    atomicAdd(yd + 128 + lane * 4 + 3, a1.w * w);
}

// --- final projection: out = h @ W + b via f32 WMMA -------------------------
// One wave computes one 16x16 output tile. 4 waves / block.
// A 16x4 f32 fragment: lane<16 -> M=lane, K={0,1}; lane>=16 -> M=lane-16, K={2,3}
// B 4x16 f32 fragment: row K striped across lanes within a VGPR (mirror of A)
// C/D 16x16 f32: VGPR v -> M = v + (lane>=16 ? 8 : 0), N = lane&15
__global__ __launch_bounds__(128)
void sgc_gemm_wmma(const float* __restrict__ h,    // [N, 256]
                   const float* __restrict__ Wm,   // [256, 64]
                   const float* __restrict__ bias, // [64]
                   float* __restrict__ out,        // [N, 64]
                   int nRows) {
    const int lane = threadIdx.x & 31;
    const int wave = threadIdx.x >> 5;
    const int tile = blockIdx.x * 4 + wave;
    const int rowTile = tile >> 2;          // N_CLASS/16 == 4 column tiles
    const int colTile = tile & 3;
    if (rowTile * 16 >= nRows) return;      // wave-uniform, EXEC stays full
    const int row0 = rowTile * 16;
    const int col0 = colTile * 16;

    const int mn = lane & 15;               // M for A, N for B/C/D
    const int kb = (lane >> 4) << 1;        // 0 or 2

    const float* aPtr = h  + (size_t)(row0 + mn) * SGC_F + kb;
    const float* bPtr = Wm + (size_t)kb * SGC_C + (col0 + mn);

    v8f acc = {};
#pragma unroll
    for (int k = 0; k < SGC_F; k += 4) {
        v2f a = *(const v2f*)(aPtr + k);    // contiguous K pair -> b64 load
        v2f b;
        b.x = bPtr[(size_t)k * SGC_C];
        b.y = bPtr[(size_t)k * SGC_C + SGC_C];
        acc = __builtin_amdgcn_wmma_f32_16x16x4_f32(
            /*neg_a=*/false, a, /*neg_b=*/false, b,
            /*c_mod=*/(short)0, acc, /*reuse_a=*/false, /*reuse_b=*/false);
    }

    const float bv   = bias[col0 + mn];
    const int   nOut = col0 + mn;
    const int   mOff = row0 + ((lane >> 4) << 3);
#pragma unroll
    for (int v = 0; v < 8; ++v)
        out[(size_t)(mOff + v) * SGC_C + nOut] = acc[v] + bv;
}

extern "C" void kernel_launch(void* const* d_in, const int* in_sizes, int n_in,
                              void* d_out, int out_size, void* d_ws, size_t ws_size,
                              hipStream_t stream) {
    const float*     x    = (const float*)d_in[0];       // [N, 256]
    const long long* ei   = (const long long*)d_in[1];   // [2, E] int64
    const float*     Wm   = (const float*)d_in[2];       // [256, 64]
    const float*     bv   = (const float*)d_in[3];       // [64]
    float*           out  = (float*)d_out;               // [N, 64]

    const int N = SGC_N, E = SGC_E, F = SGC_F;
    const long long* src = ei;
    const long long* dst = ei + E;

    // workspace: deg | dinv | bufA | bufB   (~103 MB)
    float* deg  = (float*)d_ws;
    float* dinv = deg  + N;
    float* bufA = dinv + N;                 // byte offset 400000, 16B aligned
    float* bufB = bufA + (size_t)N * F;

    sgc_deg_init <<<(N + 255) / 256, 256, 0, stream>>>(deg, N);
    sgc_deg_accum<<<(E + 255) / 256, 256, 0, stream>>>(dst, deg, E);
    sgc_dinv     <<<(N + 255) / 256, 256, 0, stream>>>(deg, dinv, N);

    const float* cur = x;
    float* nxt = bufA;
    float* other = bufB;
    for (int hop = 0; hop < 3; ++hop) {
        sgc_self_init<<<(N * F + 255) / 256, 256, 0, stream>>>(cur, dinv, nxt, N * F);
        sgc_hop_edges<<<(E + 7) / 8, 256, 0, stream>>>(cur, nxt, src, dst, dinv, E);
        cur = nxt;
        float* t = other; other = nxt; nxt = t;
    }
    // cur: bufA after hop 3 (x -> A -> B -> A)

    // 3125 row tiles x 4 col tiles = 12500 tiles; 4 waves/block -> 3125 blocks
    sgc_gemm_wmma<<<3125, 128, 0, stream>>>(cur, Wm, bv, out, N);
}